// DuhamelLayer_56650618635121
// MI455X (gfx1250) — compile-verified
//
#include <hip/hip_runtime.h>
#include <math.h>

typedef float v2f __attribute__((ext_vector_type(2)));
typedef float v4f __attribute__((ext_vector_type(4)));
typedef float v8f __attribute__((ext_vector_type(8)));
typedef unsigned int u32x4 __attribute__((ext_vector_type(4)));
typedef int i32x4 __attribute__((ext_vector_type(4)));
typedef int i32x8 __attribute__((ext_vector_type(8)));

#define OUT_CH 8
#define MAXK   3687            // max padded kernel length (odd)
#define PAD    1843            // MAXK/2
#define CK     128             // K taps per TDM chunk
#define NCHUNK 29
#define KTOT   (CK * NCHUNK)   // 3712 >= MAXK + 16, multiple of CK
#define CELEM  (CK * 16)       // floats per B chunk (2048)
#define NT     65536
#define BATCH  16
#define TILE_T 32              // output times per wave (16 direct + 16 shifted slots)
#define WAVES  4
#define BLOCK  (WAVES * 32)
#define SEG_T  (TILE_T * WAVES)     // 128 output times per workgroup
#define XS_LEN 3824            // (SEG_T-TILE_T) + 15 + 2 + (KTOT-4) + 1 = 3823 -> 3824

__device__ __constant__ int d_VW[OUT_CH] = {1844, 1317, 1024, 768, 576, 419, 307, 230};

// ---------------------------------------------------------------------------
// Kernel 1: filter bank in PAIRED layout: Bp[k>>1][n][k&1] = wpad_slot[n][k]
//   slot n<8 :  wpad[n][k]          slot n>=8 : wpad[n-8][k-16]
// so each lane's (k, k+1) B pair is contiguous -> one ds_load_b64 per WMMA.
// ---------------------------------------------------------------------------
__global__ void build_filters(const float* __restrict__ log_omegas,
                              float* __restrict__ Bp) {
  int idx = blockIdx.x * blockDim.x + threadIdx.x;
  if (idx >= KTOT * 16) return;
  int k = idx >> 4;
  int n = idx & 15;
  int c = n & 7;
  int kk = k - ((n >= 8) ? 16 : 0);

  float val = 0.0f;
  int W   = d_VW[c];
  int off = 1844 - W;
  int j   = kk - off;
  if (kk >= 0 && kk < MAXK && j >= 0 && j < W) {
    float omega  = fminf(fmaxf(expf(log_omegas[c]), 0.01f), 1000.0f);
    float sq     = sqrtf(1.0f - 0.05f * 0.05f);
    float omegaD = omega * sq;
    float tt     = (float)(W - 1 - j) * 0.01f;
    val = (1.0f / omegaD) * expf(-0.05f * omega * tt) * sinf(omegaD * tt);
  }
  Bp[(size_t)(k >> 1) * 32 + (n << 1) + (k & 1)] = val;
}

// ---------------------------------------------------------------------------
// TDM: 1D copy of one B chunk (CELEM f32) from global into LDS.
// D# per ISA ch.8: group0 = {count|flags, lds_addr, ga_lo, ga_hi|type=2},
// group1 = dims/strides, groups 2/3 zero (<=2D tensor).
// clang-23 toolchain: 6-arg builtin (u32x4, i32x8, i32x4, i32x4, i32x8, cpol).
// ---------------------------------------------------------------------------
__device__ __forceinline__ void tdm_load_chunk(const float* gsrc, void* ldsdst) {
  unsigned long long ga = (unsigned long long)(uintptr_t)gsrc;
  unsigned lds_off = (unsigned)(uintptr_t)ldsdst;   // low 32 bits = LDS byte offset

  u32x4 g0;
  g0.x = 1u;                                        // count=1, user descriptor
  g0.y = lds_off;                                   // lds_addr
  g0.z = (unsigned)(ga & 0xFFFFFFFFu);              // global_addr[31:0]
  g0.w = (unsigned)((ga >> 32) & 0x01FFFFFFu) | (2u << 30);  // [56:32] | type=2

  const int ne = CELEM;                             // elements (data_size units)
  i32x8 g1;
  g1[0] = (2 << 16);                 // wg_mask=0, data_size=2 (4B), no flags
  g1[1] = (ne & 0xFFFF) << 16;       // [63:48]  tensor_dim0[15:0]
  g1[2] = ((ne >> 16) & 0xFFFF)      // [79:64]  tensor_dim0[31:16]
        | (1 << 16);                 // [95:80]  tensor_dim1[15:0] = 1
  g1[3] = (ne & 0xFFFF) << 16;       // [111:96] dim1 hi = 0 | [127:112] tile_dim0
  g1[4] = 1;                         // [143:128] tile_dim1=1, [159:144] tile_dim2=0
  g1[5] = ne;                        // tensor_dim0_stride[31:0]
  g1[6] = (ne & 0xFFFF) << 16;       // stride0 hi16=0 | tensor_dim1_stride[15:0]
  g1[7] = (ne >> 16);                // tensor_dim1_stride[47:16]

  i32x4 z4 = {0, 0, 0, 0};
  i32x8 z8 = {0, 0, 0, 0, 0, 0, 0, 0};
  __builtin_amdgcn_tensor_load_to_lds(g0, g1, z4, z4, z8, 0);
}

// ---------------------------------------------------------------------------
// Kernel 2: implicit-GEMM conv, V_WMMA_F32_16X16X4_F32, TDM double-buffered B.
// ---------------------------------------------------------------------------
__global__ void __launch_bounds__(BLOCK)
duhamel_conv(const float* __restrict__ x,
             const float* __restrict__ Bp,
             float* __restrict__ out) {
  __shared__ float xs[XS_LEN];
  __shared__ __align__(16) float Bbuf[2][CELEM];   // 2 x 8KB double buffer

  const int seg  = blockIdx.x;
  const int b    = blockIdx.y;
  const int tid  = threadIdx.x;
  const int lane = tid & 31;
  const int wv   = tid >> 5;

  // Kick off TDM for chunk 0 immediately (overlaps with x staging below).
  if (wv == 0) tdm_load_chunk(Bp, &Bbuf[0][0]);

  // Cooperative zero-padded stage of the input segment into LDS.
  const int    gbase = seg * SEG_T - PAD;
  const float* xb    = x + (size_t)b * NT;
  for (int i = tid; i < XS_LEN; i += BLOCK) {
    int g = gbase + i;
    xs[i] = (g >= 0 && g < NT) ? xb[g] : 0.0f;
  }
  if (wv == 0) __builtin_amdgcn_s_wait_tensorcnt(0);
  __syncthreads();

  // Fragment addressing (wave32 ISA layouts):
  //  A 16x4 f32: lanes 0-15 -> M=lane, K={k,k+1}; lanes 16-31 -> M=lane-16, K={k+2,k+3}
  //  B 4x16 f32: VGPR v: lanes 0-15 -> K=v, lanes 16-31 -> K=v+2 ; N = lane&15
  const int mlo = lane & 15;
  const int kh  = (lane >> 4) << 1;           // 0 or 2
  const int ab  = wv * TILE_T + mlo + kh;     // LDS x index base
  const int bb  = (kh >> 1) * 32 + (mlo << 1);// paired-B lane offset (k-pair stride 32)

  v8f acc = {0.f, 0.f, 0.f, 0.f, 0.f, 0.f, 0.f, 0.f};

  for (int c = 0; c < NCHUNK; ++c) {
    // Prefetch next chunk into the other buffer (safe: last read of it was
    // one iteration ago, fenced by the barrier at the end of iteration c-1).
    if (wv == 0 && c + 1 < NCHUNK)
      tdm_load_chunk(Bp + (size_t)(c + 1) * CELEM, &Bbuf[(c + 1) & 1][0]);

    const float* __restrict__ Bc = &Bbuf[c & 1][0];
    const int kg = c * CK;                    // global K base of this chunk
#pragma unroll
    for (int kc = 0; kc < CK; kc += 4) {
      v2f a, bf;
      a.x = xs[ab + kg + kc];
      a.y = xs[ab + kg + kc + 1];
      bf  = *(const v2f*)(Bc + (kc << 3) + bb);   // ds_load_b64 (pair layout)
      acc = __builtin_amdgcn_wmma_f32_16x16x4_f32(
          false, a, false, bf, (short)0, acc, false, false);
    }

    if (wv == 0) __builtin_amdgcn_s_wait_tensorcnt(0);
    __syncthreads();
  }

  // D layout: VGPR v, lanes 0-15 -> M=v; lanes 16-31 -> M=v+8; N = lane&15.
  // slot n<8: (t0+M, ch=n); slot n>=8: (t0+16+M, ch=n-8) -> 8 consecutive times.
  const int ch = mlo & 7;
  const int t0 = seg * SEG_T + wv * TILE_T;
  const int t  = t0 + ((lane >> 4) << 3) + ((mlo >> 3) << 4);
  float* __restrict__ po = out + (size_t)(b * OUT_CH + ch) * NT + t;

  v4f lo = {acc[0], acc[1], acc[2], acc[3]};
  v4f hi = {acc[4], acc[5], acc[6], acc[7]};
  *(v4f*)(po)     = lo;    // 32B aligned (t ≡ 0 mod 8)
  *(v4f*)(po + 4) = hi;
}

// ---------------------------------------------------------------------------
extern "C" void kernel_launch(void* const* d_in, const int* in_sizes, int n_in,
                              void* d_out, int out_size, void* d_ws, size_t ws_size,
                              hipStream_t stream) {
  const float* x   = (const float*)d_in[0];   // [16,1,65536] f32
  const float* lw  = (const float*)d_in[1];   // [8] f32 log_omegas
  float*       out = (float*)d_out;           // [16,8,65536] f32
  float*       Bp  = (float*)d_ws;            // paired filter bank, 232 KB

  int nB = KTOT * 16;
  build_filters<<<(nB + 255) / 256, 256, 0, stream>>>(lw, Bp);

  dim3 grid(NT / SEG_T, BATCH);
  duhamel_conv<<<grid, BLOCK, 0, stream>>>(x, Bp, out);
}